// KANEdgeFunction_51539607552783
// MI455X (gfx1250) — compile-verified
//
#include <hip/hip_runtime.h>

// ---------------------------------------------------------------------------
// KAN edge function: out = w_base * silu(x) + w_spline * B-spline(x) . coeffs
// Uniform cubic B-spline, GRID_SIZE=8, DEGREE=3, domain [-5, 5], h = 1.25.
// Closed form: for x in knot cell jj (extended knots at -8.75 + 1.25*jj),
// only 4 bases are nonzero with the standard uniform cubic weights.
// Truncated boundary bases handled by a zero-padded LDS LUT.
// Data path: per-wave double-buffered GLOBAL_LOAD_ASYNC_TO_LDS_B128 pipeline
// (ASYNCcnt), compute from LDS, global_store_b128 out. ~23.3 TB/s roofline.
// ---------------------------------------------------------------------------

#if defined(__HIP_DEVICE_COMPILE__) && defined(__gfx1250__)
  #if __has_builtin(__builtin_amdgcn_global_load_async_to_lds_b128) && \
      __has_builtin(__builtin_amdgcn_s_wait_asynccnt)
    #define KAN_ASYNC 1
  #else
    #define KAN_ASYNC 0
  #endif
#else
  #define KAN_ASYNC 0
#endif

#if KAN_ASYNC
// Exact parameter types per clang diagnostic: b128 variant takes pointers to
// 16-byte int vectors in AS(1) (global) and AS(3) (LDS).
typedef int kan_v4i __attribute__((__vector_size__(4 * sizeof(int))));
typedef __attribute__((address_space(1))) kan_v4i kan_gv4i;
typedef __attribute__((address_space(3))) kan_v4i kan_lv4i;
#endif

__device__ __forceinline__ float kan_eval(float x, float wb, float ws,
                                          const float* C) {
  // silu(x) = x * sigmoid(x) = x / (1 + exp(-x)); use v_exp_f32 + v_rcp_f32
  const float LOG2E = 1.44269504088896340736f;
#ifdef __HIP_DEVICE_COMPILE__
  float sig = __builtin_amdgcn_rcpf(1.0f + __builtin_amdgcn_exp2f(x * -LOG2E));
#else
  float sig = 1.0f / (1.0f + expf(-x));
#endif
  float base = x * sig;

  // Extended-knot cell: xf = (x - (LO - 3h)) / h = (x + 8.75) * 0.8
  float xf = (x + 8.75f) * 0.8f;
  float jf = floorf(xf);
  float t  = xf - jf;                      // local coordinate in [0,1)
  bool inr = (xf >= 0.0f) && (xf < 14.0f); // inside extended support
  int  jj  = inr ? (int)jf : 0;            // jj in [0,13]

  float t2 = t * t;
  float t3 = t2 * t;
  float u  = 1.0f - t;
  const float K6 = 0.16666666666666666f;
  float w0 = u * u * u * K6;                                  // B_{jj-3}
  float w1 = (3.0f * t3 - 6.0f * t2 + 4.0f) * K6;             // B_{jj-2}
  float w2 = (-3.0f * t3 + 3.0f * t2 + 3.0f * t + 1.0f) * K6; // B_{jj-1}
  float w3 = t3 * K6;                                         // B_{jj}

  // LUT: C[k] = coeffs[k-3] for 3<=k<=13, else 0  (max index used: 16)
  float s = w0 * C[jj] + w1 * C[jj + 1] + w2 * C[jj + 2] + w3 * C[jj + 3];
  s = inr ? s : 0.0f;

  return wb * base + ws * s;
}

__global__ __launch_bounds__(256) void kan_edge_kernel(
    const float* __restrict__ x, const float* __restrict__ wbp,
    const float* __restrict__ wsp, const float* __restrict__ coef,
    float* __restrict__ out, long long n) {
  __shared__ __align__(16) float sC[32];        // zero-padded coeff LUT
  __shared__ __align__(16) float sStage[2048];  // 8 waves * 2 bufs * 128 floats

  const int tid = threadIdx.x;
  if (tid < 32) {
    float v = 0.0f;
    if (tid >= 3 && tid <= 13) v = coef[tid - 3];
    sC[tid] = v;
  }
  __syncthreads();

  const float wb = wbp[0];
  const float ws = wsp[0];

  const long long n4  = n >> 2;
  const long long gsz = (long long)gridDim.x * blockDim.x;
  const long long i0  = (long long)blockIdx.x * blockDim.x + tid;

#if KAN_ASYNC
  // Per-wave double-buffered async global->LDS pipeline. Each wave owns its
  // 2x512B buffers, so only per-wave s_wait_asynccnt is needed (no barriers).
  const int lane = tid & 31;
  float* bufA = &sStage[(tid >> 5) * 256] + lane * 4;
  float* bufB = bufA + 128;

  int cur = 0;
  if (i0 < n4) {
    __builtin_amdgcn_global_load_async_to_lds_b128(
        (kan_gv4i*)(x + 4 * i0), (kan_lv4i*)bufA, 0, 0);
  }
  for (long long j = i0; j < n4; j += gsz) {
    const long long jn = j + gsz;
    float* curp = cur ? bufB : bufA;
    float* nxtp = cur ? bufA : bufB;
    if (jn < n4) {
      __builtin_amdgcn_global_load_async_to_lds_b128(
          (kan_gv4i*)(x + 4 * jn), (kan_lv4i*)nxtp, 0, 0);
      __builtin_amdgcn_s_wait_asynccnt(1);  // previous stage landed in LDS
    } else {
      __builtin_amdgcn_s_wait_asynccnt(0);
    }
    const float4 v = *(const float4*)curp;  // ds_load_b128
    float4 r;
    r.x = kan_eval(v.x, wb, ws, sC);
    r.y = kan_eval(v.y, wb, ws, sC);
    r.z = kan_eval(v.z, wb, ws, sC);
    r.w = kan_eval(v.w, wb, ws, sC);
    *(float4*)(out + 4 * j) = r;            // global_store_b128
    cur ^= 1;
    asm volatile("" ::: "memory");          // pin issue order vs. LDS reuse
  }
#else
  for (long long j = i0; j < n4; j += gsz) {
    const float4 v = *(const float4*)(x + 4 * j);
    float4 r;
    r.x = kan_eval(v.x, wb, ws, sC);
    r.y = kan_eval(v.y, wb, ws, sC);
    r.z = kan_eval(v.z, wb, ws, sC);
    r.w = kan_eval(v.w, wb, ws, sC);
    *(float4*)(out + 4 * j) = r;
  }
#endif

  // Scalar tail (n not divisible by 4) -- no-op for the 8192x2048 harness.
  if (blockIdx.x == 0 && tid == 0) {
    for (long long j = n4 << 2; j < n; ++j)
      out[j] = kan_eval(x[j], wb, ws, sC);
  }
}

extern "C" void kernel_launch(void* const* d_in, const int* in_sizes, int n_in,
                              void* d_out, int out_size, void* d_ws,
                              size_t ws_size, hipStream_t stream) {
  const float* x  = (const float*)d_in[0];  // [N, D] fp32
  const float* wb = (const float*)d_in[1];  // [1]
  const float* ws = (const float*)d_in[2];  // [1]
  const float* cf = (const float*)d_in[3];  // [11]
  float* out = (float*)d_out;

  const long long n  = (long long)in_sizes[0];
  const long long n4 = n >> 2;

  const int threads = 256;
  long long blocks = (n4 + threads - 1) / threads;
  if (blocks > 2048) blocks = 2048;  // 8 grid-stride stages per thread at full size
  if (blocks < 1) blocks = 1;

  kan_edge_kernel<<<(int)blocks, threads, 0, stream>>>(x, wb, ws, cf, out, n);
}